// GtNet_63118839382155
// MI455X (gfx1250) — compile-verified
//
#include <hip/hip_runtime.h>
#include <stdint.h>

#define B_    32
#define H_    256
#define W_    256
#define HW    (H_*W_)
#define NC    25
#define KW    5

#define TILE  32
#define TDIMX 32
#define TDIMY 8
#define NTHR  (TDIMX*TDIMY)
#define ROWS_PER_THREAD 4
#define GT_DIM 40   // TILE + 2*4 (appear halo 2 needs gt halo 2 more)
#define AP_DIM 36   // TILE + 2*2

typedef __attribute__((ext_vector_type(4))) unsigned int v4u;
typedef __attribute__((ext_vector_type(8))) int          v8i;
typedef __attribute__((ext_vector_type(4))) int          v4i;

__global__ __launch_bounds__(NTHR)
void gtnet_fused(const float* __restrict__ im_input,
                 const float* __restrict__ m_kernel,
                 const int*   __restrict__ gt_motion,
                 float* __restrict__ out)
{
    __shared__ float kbuf[(NC+1)*NC];          // 26 rows x 25; row 25 = zeros (OOB sentinel)
    __shared__ int   gt_lds[GT_DIM*GT_DIM];
    __shared__ float ap_lds[AP_DIM*AP_DIM];
    __shared__ float im_lds[3][AP_DIM*AP_DIM];

    const int b   = blockIdx.z;
    const int y0  = blockIdx.y * TILE;
    const int x0  = blockIdx.x * TILE;
    const int tid = threadIdx.y * TDIMX + threadIdx.x;

    // ---- TDM: DMA the 25x25 kernel table (625 f32, contiguous) into LDS ----
    if (tid == 0) {
        uint64_t ga = (uint64_t)(uintptr_t)m_kernel;
        uint32_t lds_addr = (uint32_t)(uintptr_t)(&kbuf[0]);   // LDS aperture: low 32 bits = byte offset
        v4u g0;
        g0[0] = 1u;                                     // count=1, user descriptor, no gather
        g0[1] = lds_addr;                               // lds_addr [63:32]
        g0[2] = (uint32_t)ga;                           // global_addr[31:0]
        g0[3] = ((uint32_t)(ga >> 32) & 0x1FFFFFFu)     // global_addr[56:32]
                | (2u << 30);                           // type=2 ("image")
        v8i g1;
        g1[0] = (int)(2u   << 16);                      // data_size = 4 bytes
        g1[1] = (int)(625u << 16);                      // tensor_dim0[15:0] = 625
        g1[2] = (int)(1u   << 16);                      // tensor_dim0 hi=0, tensor_dim1=1
        g1[3] = (int)(625u << 16);                      // tile_dim0 = 625
        g1[4] = 0;                                      // tile_dim1=0 (1-D), tile_dim2=0
        g1[5] = 625;                                    // tensor_dim0_stride low
        g1[6] = 0;                                      // stride hi=0, tensor_dim1_stride=0
        g1[7] = 0;
        v4i g2 = {0,0,0,0};                             // group 2 unused (<=2-D tensor)
        v4i g3 = {0,0,0,0};                             // group 3 unused
        v8i g4 = {0,0,0,0,0,0,0,0};                     // extra group (6-arg toolchain form)
        __builtin_amdgcn_tensor_load_to_lds(g0, g1, g2, g3, g4, 0);
    }
    // zero sentinel class row -> zero weight for out-of-image taps (zero-padded conv)
    if (tid < NC) kbuf[NC*NC + tid] = 0.0f;

    // ---- stage gt tile + halo(4); OOB -> sentinel class 25 ----
    const int* gtb = gt_motion + b * HW;
    for (int i = tid; i < GT_DIM*GT_DIM; i += NTHR) {
        int ly = i / GT_DIM, lx = i - ly*GT_DIM;
        int gy = y0 - 4 + ly, gx = x0 - 4 + lx;
        int v = NC;
        if (gy >= 0 && gy < H_ && gx >= 0 && gx < W_) v = gtb[gy * W_ + gx];
        gt_lds[i] = v;
    }

    // ---- stage im (last 3 of 6 channels) tile + halo(2); OOB -> 0 ----
    const float* imb = im_input + (size_t)b * 6 * HW + (size_t)3 * HW;
    for (int i = tid; i < AP_DIM*AP_DIM; i += NTHR) {
        int ly = i / AP_DIM, lx = i - ly*AP_DIM;
        int gy = y0 - 2 + ly, gx = x0 - 2 + lx;
        float v0 = 0.f, v1 = 0.f, v2 = 0.f;
        if (gy >= 0 && gy < H_ && gx >= 0 && gx < W_) {
            int o = gy * W_ + gx;
            v0 = imb[o]; v1 = imb[HW + o]; v2 = imb[2*HW + o];
        }
        im_lds[0][i] = v0; im_lds[1][i] = v1; im_lds[2][i] = v2;
    }

    __builtin_amdgcn_s_wait_tensorcnt(0);
    __syncthreads();

    float* out_pred = out;                                        // [B,3,H,W]
    float* out_mask = out + (size_t)B_*3*HW;                      // [B,25,H,W]
    float* out_dis  = out + (size_t)B_*3*HW + (size_t)B_*NC*HW;   // [B,1,H,W]

    // ---- appear over 36x36 (tile + halo 2); emit 1-appear for central pixels ----
    for (int i = tid; i < AP_DIM*AP_DIM; i += NTHR) {
        int ly = i / AP_DIM, lx = i - ly*AP_DIM;
        float seg = 0.f;
        #pragma unroll
        for (int ky = 0; ky < KW; ++ky)
            #pragma unroll
            for (int kx = 0; kx < KW; ++kx)
                seg += kbuf[gt_lds[(ly+ky)*GT_DIM + (lx+kx)] * NC + (ky*KW + kx)];
        float d  = seg - 1.f; d = d > 0.f ? d : 0.f;    // disappear = relu(seg-1)
        float ap = 1.f - d;   ap = ap > 0.f ? ap : 0.f; // appear = relu(1-disappear)
        ap_lds[i] = ap;
        if (ly >= 2 && ly < 2 + TILE && lx >= 2 && lx < 2 + TILE)
            out_dis[(size_t)b*HW + (size_t)(y0 + ly - 2)*W_ + (x0 + lx - 2)] = 1.f - ap;
    }
    __syncthreads();

    // ---- per-pixel: one-hot mask stream + 25-tap gather conv for pred ----
    const int tx = threadIdx.x;
    for (int r = 0; r < ROWS_PER_THREAD; ++r) {
        const int ty = threadIdx.y + r * TDIMY;
        const int gy = y0 + ty, gx = x0 + tx;

        const int g = gt_lds[(ty + 4)*GT_DIM + (tx + 4)];
        const size_t mbase = ((size_t)b*NC)*HW + (size_t)gy*W_ + gx;
        #pragma unroll
        for (int k = 0; k < NC; ++k)
            out_mask[mbase + (size_t)k*HW] = (k == g) ? 1.0f : 0.0f;

        float p0 = 0.f, p1 = 0.f, p2 = 0.f;
        #pragma unroll
        for (int ky = 0; ky < KW; ++ky) {
            #pragma unroll
            for (int kx = 0; kx < KW; ++kx) {
                const int ai = (ty + ky)*AP_DIM + (tx + kx);
                const float w =
                    kbuf[gt_lds[(ty + ky + 2)*GT_DIM + (tx + kx + 2)] * NC + (ky*KW + kx)]
                    * ap_lds[ai];
                p0 += w * im_lds[0][ai];
                p1 += w * im_lds[1][ai];
                p2 += w * im_lds[2][ai];
            }
        }
        const size_t pbase = ((size_t)b*3)*HW + (size_t)gy*W_ + gx;
        out_pred[pbase]        = p0;
        out_pred[pbase +   HW] = p1;
        out_pred[pbase + 2*HW] = p2;
    }
}

extern "C" void kernel_launch(void* const* d_in, const int* in_sizes, int n_in,
                              void* d_out, int out_size, void* d_ws, size_t ws_size,
                              hipStream_t stream) {
    (void)in_sizes; (void)n_in; (void)out_size; (void)d_ws; (void)ws_size;
    const float* im = (const float*)d_in[0];   // [32,6,256,256] f32
    const float* kn = (const float*)d_in[1];   // [1,25,5,5]     f32
    const int*   gt = (const int*)d_in[2];     // [32,1,256,256] i32
    dim3 grid(W_/TILE, H_/TILE, B_);           // 8 x 8 x 32
    dim3 block(TDIMX, TDIMY);                  // 256 threads = 8 waves
    hipLaunchKernelGGL(gtnet_fused, grid, block, 0, stream, im, kn, gt, (float*)d_out);
}